// DWT_Pooling_60713657696827
// MI455X (gfx1250) — compile-verified
//
#include <hip/hip_runtime.h>

// DWT_Pooling (bior1.3 LL band) == scaled 2x2 average pool.
// x: [16,64,256,256] f32  ->  out: [16,64,128,128] f32
// Memory-bound: 320 MB traffic -> ~13.7 us floor @ 23.3 TB/s.
// Compute path uses exact-fp32 V_WMMA_F32_16X16X4_F32 per 16x16 output tile.

typedef __attribute__((ext_vector_type(2))) float v2f;
typedef __attribute__((ext_vector_type(8))) float v8f;

#define CDWT 0.70710678118654752f

#define H       256
#define W       256
#define HO      128
#define WO      128
#define IMG_IN  (H * W)      // 65536
#define IMG_OUT (HO * WO)    // 16384
#define LSTRIDE 34           // LDS row stride (floats): conflict-free fragment reads

__global__ __launch_bounds__(256) void dwt_ll_wmma(const float* __restrict__ x,
                                                   float* __restrict__ out) {
    // Per-wave private 32x32 f32 input tile, padded rows (32 * 34 floats).
    __shared__ float lds[8][32 * LSTRIDE];

    const int tid  = threadIdx.x;
    const int wv   = tid >> 5;
    const int lane = tid & 31;

    // 8x8 output tiles per image, 1024 images -> 65536 waves total.
    const int waveId = blockIdx.x * 8 + wv;
    const int tx  = waveId & 7;
    const int ty  = (waveId >> 3) & 7;
    const int img = waveId >> 6;

    const float* __restrict__ src = x   + (size_t)img * IMG_IN  + (ty * 32) * W  + tx * 32;
    float*       __restrict__ dst = out + (size_t)img * IMG_OUT + (ty * 16) * WO + tx * 16;

    // ---- Stage 1: coalesced global -> LDS (32x32 tile) -------------------
    // Lanes 0..7 cover one 128B row segment; 4 rows per iteration, 8 iters.
    {
        const int q     = lane & 7;   // float4 column index within row
        const int rbase = lane >> 3;  // 0..3
        float* lp = &lds[wv][0];
#pragma unroll
        for (int i = 0; i < 8; ++i) {
            const int r = i * 4 + rbase;
            const float4 v = *(const float4*)(src + r * W + q * 4);
            float* p = lp + r * LSTRIDE + q * 4;
            v2f lo, hic;
            lo.x = v.x; lo.y = v.y;
            hic.x = v.z; hic.y = v.w;
            *(v2f*)(p)     = lo;   // 8B-aligned ds_store_b64
            *(v2f*)(p + 2) = hic;
        }
    }
    __syncthreads();  // cheap; guarantees cross-lane LDS visibility

    // ---- Stage 2: vertical pair-sums (VALU) + horizontal pair-sums (WMMA)
    // A fragment (16x4 f32): lane L -> row m = L&15; VGPR0 = K 4k+2hi, VGPR1 = K 4k+2hi+1.
    // B fragment (4x16 f32): constant pair-sum matrix; both components = c*(n == 2k+hi).
    const int m  = lane & 15;
    const int hi = lane >> 4;

    v8f acc = {};  // C = 0
    const float* sp = &lds[wv][0];
#pragma unroll
    for (int k = 0; k < 8; ++k) {
        const int c = 4 * k + 2 * hi;
        const v2f p0 = *(const v2f*)(sp + (2 * m)     * LSTRIDE + c);
        const v2f p1 = *(const v2f*)(sp + (2 * m + 1) * LSTRIDE + c);
        v2f a;
        a.x = CDWT * p0.x + CDWT * p1.x;  // Lrows rounding mirrors reference
        a.y = CDWT * p0.y + CDWT * p1.y;
        const float bval = (m == 2 * k + hi) ? CDWT : 0.0f;
        v2f b;
        b.x = bval;
        b.y = bval;
        // D = A(16x4) * B(4x16) + C : accumulates LL[m,n] = c*(S[m,2n]+S[m,2n+1])
        acc = __builtin_amdgcn_wmma_f32_16x16x4_f32(
            /*neg_a=*/false, a, /*neg_b=*/false, b,
            /*c_mod=*/(short)0, acc, /*reuse_a=*/false, /*reuse_b=*/false);
    }

    // ---- Stage 3: store D fragment (VGPR v -> row v+8*hi, col n=m) -------
#pragma unroll
    for (int v = 0; v < 8; ++v) {
        dst[(v + 8 * hi) * WO + m] = acc[v];
    }
}

extern "C" void kernel_launch(void* const* d_in, const int* in_sizes, int n_in,
                              void* d_out, int out_size, void* d_ws, size_t ws_size,
                              hipStream_t stream) {
    const float* x = (const float*)d_in[0];
    float* out = (float*)d_out;
    // 65536 output tiles / 8 waves per 256-thread block -> 8192 blocks, exact cover.
    dwt_ll_wmma<<<8192, 256, 0, stream>>>(x, out);
}